// GraphBased_73315091742787
// MI455X (gfx1250) — compile-verified
//
#include <hip/hip_runtime.h>
#include <hip/hip_bf16.h>
#include <math.h>

typedef float v2f __attribute__((ext_vector_type(2)));
typedef float v8f __attribute__((ext_vector_type(8)));

#define LDIM 500
#define NBINS 4096

__device__ __forceinline__ float lrelu_f(float v) { return v > 0.f ? v : 0.01f * v; }

// ---------------------------------------------------------------------------
// f32 WMMA GEMM, LDS-staged: C[M,N] = A[M,K] * op(B) (+ bias) (+ C if beta)
// TRANSB=1: B is [N,K] row-major (B^T used). TRANSB=0: B is [K,N] row-major.
// Block = 128 threads (4 waves); block tile 64(M) x 16(N); K staged in chunks
// of 32 through LDS; each wave owns one 16x16 C tile and runs 8 WMMAs/chunk.
// Requires: M % 64 == 0, K % 4 == 0, lda/ldb % 4 == 0 (true for all calls).
// ---------------------------------------------------------------------------
#define KC 32
template <int TRANSB>
__global__ void wmma_gemm_f32(const float* __restrict__ A, const float* __restrict__ B,
                              float* __restrict__ C,
                              int N, int K, int lda, int ldb, int ldc,
                              const float* __restrict__ bias, int beta, int act)
{
    __shared__ float As[64][KC + 1];   // +1 pad: rows hit distinct banks
    __shared__ float Bs[KC][16];

    const int tid  = threadIdx.x;
    const int lane = tid & 31;
    const int wave = tid >> 5;
    const int tm   = blockIdx.y * 64;
    const int tn   = blockIdx.x * 16;
    const int hs   = lane >> 4;        // K-half select (lanes 16-31 -> +2)
    const int l16  = lane & 15;

    const int colB  = tn + l16;
    const bool colOK = colB < N;
    const int colC  = colOK ? colB : 0;
    const int arow  = wave * 16 + l16; // this wave's row inside the A tile

    v8f acc;
#pragma unroll
    for (int r = 0; r < 8; ++r) acc[r] = 0.0f;
    if (beta) {
#pragma unroll
        for (int r = 0; r < 8; ++r) {
            int mrow = tm + wave * 16 + r + 8 * hs;
            float v = C[(long)mrow * ldc + colC];
            acc[r] = colOK ? v : 0.0f;
        }
    }

    // cooperative-load coordinates
    const int a_r = tid >> 3;          // 0..15 (A row group; +16,+32,+48)
    const int a_k = (tid & 7) * 4;     // 0..28 (float4 K offset)

    for (int k0 = 0; k0 < K; k0 += KC) {
        __syncthreads();
        // ---- stage A tile: 64 x 32, float4-coalesced ----
#pragma unroll
        for (int rr = 0; rr < 4; ++rr) {
            int row = a_r + rr * 16;
            float4 v = make_float4(0.f, 0.f, 0.f, 0.f);
            if (k0 + a_k < K)          // K % 4 == 0 -> float4 all-or-nothing
                v = *(const float4*)(A + (long)(tm + row) * lda + k0 + a_k);
            As[row][a_k + 0] = v.x;
            As[row][a_k + 1] = v.y;
            As[row][a_k + 2] = v.z;
            As[row][a_k + 3] = v.w;
        }
        // ---- stage B tile: 32 x 16 ----
        if (TRANSB) {                  // B[N,K]: Bs[kk][nn] = B[(tn+nn)*ldb + k0+kk]
            int nn = tid >> 3;         // 0..15
            int kk = (tid & 7) * 4;    // 0..28
            float4 v = make_float4(0.f, 0.f, 0.f, 0.f);
            if ((tn + nn) < N && (k0 + kk) < K)
                v = *(const float4*)(B + (long)(tn + nn) * ldb + k0 + kk);
            Bs[kk + 0][nn] = v.x;
            Bs[kk + 1][nn] = v.y;
            Bs[kk + 2][nn] = v.z;
            Bs[kk + 3][nn] = v.w;
        } else {                       // B[K,N]: Bs[kk][nn] = B[(k0+kk)*ldb + tn+nn]
            int nn = tid & 15;
            int kb = tid >> 4;         // 0..7
#pragma unroll
            for (int rr = 0; rr < 4; ++rr) {
                int kk = kb + rr * 8;
                float v = 0.f;
                if ((k0 + kk) < K && (tn + nn) < N)
                    v = B[(long)(k0 + kk) * ldb + tn + nn];
                Bs[kk][nn] = v;
            }
        }
        __syncthreads();
        // ---- 8 WMMAs over the staged chunk ----
#pragma unroll
        for (int kk = 0; kk < KC; kk += 4) {
            v2f a, b;
            a[0] = As[arow][kk + 2 * hs];
            a[1] = As[arow][kk + 2 * hs + 1];
            b[0] = Bs[kk + 2 * hs][l16];
            b[1] = Bs[kk + 2 * hs + 1][l16];
            acc = __builtin_amdgcn_wmma_f32_16x16x4_f32(false, a, false, b,
                                                        (short)0, acc, false, false);
        }
    }

    float bv = (bias && colOK) ? bias[colC] : 0.0f;
#pragma unroll
    for (int r = 0; r < 8; ++r) {
        int mrow = tm + wave * 16 + r + 8 * hs;
        float v = acc[r] + bv;
        if (act) v = v > 0.f ? v : 0.f;
        if (colOK) C[(long)mrow * ldc + colB] = v;
    }
}

// ---------------------------------------------------------------------------
// conv1 (1->20, 5x5, valid) + relu + 2x2 maxpool : [n,1,28,28] -> [n,20,12,12]
// ---------------------------------------------------------------------------
__global__ void conv1_pool(const float* __restrict__ x, const float* __restrict__ w,
                           const float* __restrict__ b, float* __restrict__ out)
{
    __shared__ float img[784];
    __shared__ float wt[500];
    __shared__ float bs[20];
    int n = blockIdx.x;
    const float* src = x + (long)n * 784;
    for (int i = threadIdx.x; i < 784; i += blockDim.x) img[i] = src[i];
    for (int i = threadIdx.x; i < 500; i += blockDim.x) wt[i] = w[i];
    if (threadIdx.x < 20) bs[threadIdx.x] = b[threadIdx.x];
    __syncthreads();
    for (int o = threadIdx.x; o < 2880; o += blockDim.x) {
        int oc = o / 144, rem = o % 144, pr = rem / 12, pc = rem % 12;
        const float* wp = wt + oc * 25;
        float best = -1e30f;
        for (int dr = 0; dr < 2; ++dr)
            for (int dc = 0; dc < 2; ++dc) {
                int r = 2 * pr + dr, c = 2 * pc + dc;
                float s = bs[oc];
                for (int i = 0; i < 5; ++i) {
                    const float* ip = img + (r + i) * 28 + c;
                    const float* wq = wp + i * 5;
#pragma unroll
                    for (int j = 0; j < 5; ++j) s += ip[j] * wq[j];
                }
                s = s > 0.f ? s : 0.f;
                best = fmaxf(best, s);
            }
        out[(long)n * 2880 + o] = best;
    }
}

// ---------------------------------------------------------------------------
// conv2 (20->50, 5x5) + relu + 2x2 maxpool : [n,20,12,12] -> [n,50,4,4] flat
// ---------------------------------------------------------------------------
__global__ void conv2_pool(const float* __restrict__ in, const float* __restrict__ w,
                           const float* __restrict__ b, float* __restrict__ out)
{
    __shared__ float img[2880];
    __shared__ float wt[5000];
    int n = blockIdx.x;
    const float* src = in + (long)n * 2880;
    for (int i = threadIdx.x; i < 2880; i += blockDim.x) img[i] = src[i];
    for (int oc0 = 0; oc0 < 50; oc0 += 10) {
        __syncthreads();
        for (int i = threadIdx.x; i < 5000; i += blockDim.x) wt[i] = w[oc0 * 500 + i];
        __syncthreads();
        for (int o = threadIdx.x; o < 160; o += blockDim.x) {
            int ocl = o / 16, rem = o % 16, pr = rem / 4, pc = rem % 4;
            int oc = oc0 + ocl;
            const float* wp = wt + ocl * 500;
            float best = -1e30f;
            for (int dr = 0; dr < 2; ++dr)
                for (int dc = 0; dc < 2; ++dc) {
                    int r = 2 * pr + dr, c = 2 * pc + dc;
                    float s = b[oc];
                    for (int ic = 0; ic < 20; ++ic) {
                        const float* ip = img + ic * 144 + r * 12 + c;
                        const float* wq = wp + ic * 25;
#pragma unroll
                        for (int i = 0; i < 5; ++i)
#pragma unroll
                            for (int j = 0; j < 5; ++j) s += ip[i * 12 + j] * wq[i * 5 + j];
                    }
                    s = s > 0.f ? s : 0.f;
                    best = fmaxf(best, s);
                }
            out[(long)n * 800 + oc * 16 + rem] = best;
        }
    }
}

__global__ void rowsq(const float* __restrict__ H, float* __restrict__ sq, int n)
{
    int i = blockIdx.x * blockDim.x + threadIdx.x;
    if (i < n) {
        float s = 0.f;
        const float* h = H + (long)i * LDIM;
        for (int k = 0; k < LDIM; ++k) s += h[k] * h[k];
        sq[i] = s;
    }
}

__global__ void zero_u32(unsigned int* p, int n)
{
    int i = blockIdx.x * blockDim.x + threadIdx.x;
    if (i < n) p[i] = 0u;
}

// histogram of chamfer distances D = 2*max(sq_i+sq_j-2G, 0), bins of width 0.5
__global__ void hist_kernel(const float* __restrict__ G, const float* __restrict__ sq,
                            unsigned int* __restrict__ hist, int n)
{
    __shared__ unsigned int lh[NBINS];
    for (int i = threadIdx.x; i < NBINS; i += blockDim.x) lh[i] = 0u;
    __syncthreads();
    long total = (long)n * n;
    for (long idx = (long)blockIdx.x * blockDim.x + threadIdx.x; idx < total;
         idx += (long)gridDim.x * blockDim.x) {
        int i = (int)(idx / n), j = (int)(idx % n);
        if (i != j) {
            float d = 2.0f * fmaxf(sq[i] + sq[j] - 2.0f * G[idx], 0.0f);
            int bin;
            if (d < 3.5f) bin = 0;
            else {
                bin = 1 + (int)floorf((d - 3.5f) * 2.0f);
                if (bin > NBINS - 1) bin = NBINS - 1;
            }
            atomicAdd(&lh[bin], 1u);
        }
    }
    __syncthreads();
    for (int i = threadIdx.x; i < NBINS; i += blockDim.x)
        if (lh[i]) atomicAdd(&hist[i], lh[i]);
}

// smallest grid threshold t_k = 3.5 + 0.5k with count(D < t_k) >= m
__global__ void thr_kernel(const unsigned int* __restrict__ hist, float* __restrict__ thr, int m)
{
    if (threadIdx.x == 0 && blockIdx.x == 0) {
        unsigned long long cum = 0;
        int k = NBINS - 1;
        for (int i = 0; i < NBINS; ++i) {
            cum += hist[i];
            if (cum >= (unsigned long long)m) { k = i; break; }
        }
        *thr = 3.5f + 0.5f * (float)k;
    }
}

// A[i,j] = (i!=j && D<thr), in-place over G; deg[i] = max(rowsum, 1)
__global__ void build_adj(float* __restrict__ G, const float* __restrict__ sq,
                          const float* __restrict__ thr, float* __restrict__ deg, int n)
{
    __shared__ float red[256];
    int i = blockIdx.x;
    float t = *thr;
    float sqi = sq[i];
    float s = 0.f;
    for (int j = threadIdx.x; j < n; j += blockDim.x) {
        float d = 2.0f * fmaxf(sqi + sq[j] - 2.0f * G[(long)i * n + j], 0.0f);
        float a = (i != j && d < t) ? 1.0f : 0.0f;
        G[(long)i * n + j] = a;
        s += a;
    }
    red[threadIdx.x] = s; __syncthreads();
    for (int st = 128; st > 0; st >>= 1) {
        if (threadIdx.x < st) red[threadIdx.x] += red[threadIdx.x + st];
        __syncthreads();
    }
    if (threadIdx.x == 0) deg[i] = fmaxf(red[0], 1.0f);
}

// X[i,:] /= deg[i]   (in place; X is [n, LDIM])
__global__ void scale_rows(float* __restrict__ X, const float* __restrict__ deg, long total)
{
    long idx = (long)blockIdx.x * blockDim.x + threadIdx.x;
    if (idx < total) X[idx] = X[idx] / deg[idx / LDIM];
}

// per-column mean / biased var of lrelu(X), X is [n, c]
__global__ void colstats(const float* __restrict__ X, float* __restrict__ mean,
                         float* __restrict__ var, int n, int c)
{
    __shared__ float rs[256], rq[256];
    int col = blockIdx.x;
    float s = 0.f, q = 0.f;
    for (int i = threadIdx.x; i < n; i += blockDim.x) {
        float v = lrelu_f(X[(long)i * c + col]);
        s += v; q += v * v;
    }
    rs[threadIdx.x] = s; rq[threadIdx.x] = q; __syncthreads();
    for (int st = 128; st > 0; st >>= 1) {
        if (threadIdx.x < st) { rs[threadIdx.x] += rs[threadIdx.x + st]; rq[threadIdx.x] += rq[threadIdx.x + st]; }
        __syncthreads();
    }
    if (threadIdx.x == 0) {
        float m = rs[0] / (float)n;
        mean[col] = m;
        var[col] = rq[0] / (float)n - m * m;
    }
}

// X = (lrelu(X) - mean)*rsqrt(var+eps)*g + b, in place, X is [n, c]
__global__ void bn_apply(float* __restrict__ X, const float* __restrict__ mean,
                         const float* __restrict__ var, const float* __restrict__ g,
                         const float* __restrict__ b, long total, int c)
{
    long idx = (long)blockIdx.x * blockDim.x + threadIdx.x;
    if (idx < total) {
        int col = (int)(idx % c);
        float v = lrelu_f(X[idx]);
        X[idx] = (v - mean[col]) * rsqrtf(var[col] + 1e-5f) * g[col] + b[col];
    }
}

// Spre[i,:] = AGG[i]·wrelᵀ + H[i]·wrootᵀ + brel  (N=2 output, weights in LDS)
__global__ void s_pre(const float* __restrict__ AGG, const float* __restrict__ H,
                      const float* __restrict__ wrel, const float* __restrict__ brel,
                      const float* __restrict__ wroot, float* __restrict__ S, int n)
{
    __shared__ float w[4 * LDIM];
    for (int i = threadIdx.x; i < 2 * LDIM; i += blockDim.x) {
        w[i] = wrel[i];
        w[2 * LDIM + i] = wroot[i];
    }
    __syncthreads();
    int i = blockIdx.x * blockDim.x + threadIdx.x;
    if (i < n) {
        float s0 = brel[0], s1 = brel[1];
        const float* a = AGG + (long)i * LDIM;
        const float* h = H + (long)i * LDIM;
        for (int k = 0; k < LDIM; ++k) {
            s0 += a[k] * w[k]        + h[k] * w[2 * LDIM + k];
            s1 += a[k] * w[LDIM + k] + h[k] * w[3 * LDIM + k];
        }
        S[2 * i] = s0; S[2 * i + 1] = s1;
    }
}

// BN (lrelu) + double softmax over the 2 columns, in place on S [n,2]
__global__ void s_final(float* __restrict__ S, const float* __restrict__ mean,
                        const float* __restrict__ var, const float* __restrict__ g,
                        const float* __restrict__ b, int n)
{
    int i = blockIdx.x * blockDim.x + threadIdx.x;
    if (i < n) {
        float v0 = lrelu_f(S[2 * i]), v1 = lrelu_f(S[2 * i + 1]);
        v0 = (v0 - mean[0]) * rsqrtf(var[0] + 1e-5f) * g[0] + b[0];
        v1 = (v1 - mean[1]) * rsqrtf(var[1] + 1e-5f) * g[1] + b[1];
        float mx = fmaxf(v0, v1);
        float e0 = __expf(v0 - mx), e1 = __expf(v1 - mx);
        float is = 1.f / (e0 + e1); v0 = e0 * is; v1 = e1 * is;
        mx = fmaxf(v0, v1);
        e0 = __expf(v0 - mx); e1 = __expf(v1 - mx);
        is = 1.f / (e0 + e1);
        S[2 * i] = e0 * is; S[2 * i + 1] = e1 * is;
    }
}

// U = A @ S   ([n,n] x [n,2]); one block per row
__global__ void a_times_s(const float* __restrict__ A, const float* __restrict__ S,
                          float* __restrict__ U, int n)
{
    __shared__ float r0[256], r1[256];
    int i = blockIdx.x;
    float s0 = 0.f, s1 = 0.f;
    for (int j = threadIdx.x; j < n; j += blockDim.x) {
        float a = A[(long)i * n + j];
        s0 += a * S[2 * j]; s1 += a * S[2 * j + 1];
    }
    r0[threadIdx.x] = s0; r1[threadIdx.x] = s1; __syncthreads();
    for (int st = 128; st > 0; st >>= 1) {
        if (threadIdx.x < st) { r0[threadIdx.x] += r0[threadIdx.x + st]; r1[threadIdx.x] += r1[threadIdx.x + st]; }
        __syncthreads();
    }
    if (threadIdx.x == 0) { U[2 * i] = r0[0]; U[2 * i + 1] = r1[0]; }
}

// Xp[c,col] = sum_i S[i,c]*Z[i,col]; one block per column
__global__ void pool_x(const float* __restrict__ S, const float* __restrict__ Z,
                       float* __restrict__ Xp, int n)
{
    __shared__ float r0[256], r1[256];
    int col = blockIdx.x;
    float s0 = 0.f, s1 = 0.f;
    for (int i = threadIdx.x; i < n; i += blockDim.x) {
        float z = Z[(long)i * LDIM + col];
        s0 += S[2 * i] * z; s1 += S[2 * i + 1] * z;
    }
    r0[threadIdx.x] = s0; r1[threadIdx.x] = s1; __syncthreads();
    for (int st = 128; st > 0; st >>= 1) {
        if (threadIdx.x < st) { r0[threadIdx.x] += r0[threadIdx.x + st]; r1[threadIdx.x] += r1[threadIdx.x + st]; }
        __syncthreads();
    }
    if (threadIdx.x == 0) { Xp[col] = r0[0]; Xp[LDIM + col] = r1[0]; }
}

// Ap[c1,c2] = sum_i S[i,c1]*U[i,c2]; single block
__global__ void pool_a(const float* __restrict__ S, const float* __restrict__ U,
                       float* __restrict__ Ap, int n)
{
    __shared__ float r[4][256];
    float a00 = 0.f, a01 = 0.f, a10 = 0.f, a11 = 0.f;
    for (int i = threadIdx.x; i < n; i += blockDim.x) {
        float s0 = S[2 * i], s1 = S[2 * i + 1];
        float u0 = U[2 * i], u1 = U[2 * i + 1];
        a00 += s0 * u0; a01 += s0 * u1; a10 += s1 * u0; a11 += s1 * u1;
    }
    r[0][threadIdx.x] = a00; r[1][threadIdx.x] = a01;
    r[2][threadIdx.x] = a10; r[3][threadIdx.x] = a11;
    __syncthreads();
    for (int st = 128; st > 0; st >>= 1) {
        if (threadIdx.x < st)
            for (int q = 0; q < 4; ++q) r[q][threadIdx.x] += r[q][threadIdx.x + st];
        __syncthreads();
    }
    if (threadIdx.x == 0)
        for (int q = 0; q < 4; ++q) Ap[q] = r[q][0];
}

// pooled SAGE + BN + lin1 + lin2 head; single block
__global__ void head_kernel(const float* __restrict__ Xp, const float* __restrict__ Ap,
                            const float* __restrict__ wrel, const float* __restrict__ brel,
                            const float* __restrict__ wroot,
                            const float* __restrict__ g1, const float* __restrict__ b1,
                            const float* __restrict__ lin1_w, const float* __restrict__ lin1_b,
                            const float* __restrict__ lin2_w, const float* __restrict__ lin2_b,
                            float* __restrict__ out, int n_out)
{
    __shared__ float xp[2 * LDIM];
    __shared__ float ag[2 * LDIM];
    __shared__ float xb[2 * LDIM];
    __shared__ float y1[LDIM];
    __shared__ float red[512];
    __shared__ float apv[4];
    int t = threadIdx.x;
    for (int i = t; i < 2 * LDIM; i += blockDim.x) xp[i] = Xp[i];
    if (t < 4) apv[t] = Ap[t];
    __syncthreads();
    float d0 = fmaxf(apv[0] + apv[1], 1.f), d1 = fmaxf(apv[2] + apv[3], 1.f);
    for (int i = t; i < 2 * LDIM; i += blockDim.x) {
        int c = i / LDIM, k = i % LDIM;
        float num = (c == 0) ? (apv[0] * xp[k] + apv[1] * xp[LDIM + k])
                             : (apv[2] * xp[k] + apv[3] * xp[LDIM + k]);
        ag[i] = num / (c == 0 ? d0 : d1);
    }
    __syncthreads();
    for (int i = t; i < 2 * LDIM; i += blockDim.x) {
        int c = i / LDIM, l = i % LDIM;
        float s = brel[l];
        const float* wr = wrel + (long)l * LDIM;
        const float* wo = wroot + (long)l * LDIM;
        const float* a = ag + c * LDIM;
        const float* x = xp + c * LDIM;
        for (int k = 0; k < LDIM; ++k) s += a[k] * wr[k] + x[k] * wo[k];
        xb[i] = lrelu_f(s);
    }
    __syncthreads();
    for (int l = t; l < LDIM; l += blockDim.x) {   // BN over the 2 pooled rows
        float v0 = xb[l], v1 = xb[LDIM + l];
        float m = 0.5f * (v0 + v1);
        float var = 0.5f * (v0 * v0 + v1 * v1) - m * m;
        float is = rsqrtf(var + 1e-5f);
        xb[l]        = (v0 - m) * is * g1[l] + b1[l];
        xb[LDIM + l] = (v1 - m) * is * g1[l] + b1[l];
    }
    __syncthreads();
    for (int j = t; j < LDIM; j += blockDim.x) {   // lin1 [1000]->[500]
        float s = lin1_b[j];
        const float* w = lin1_w + (long)j * (2 * LDIM);
        for (int k = 0; k < 2 * LDIM; ++k) s += xb[k] * w[k];
        y1[j] = lrelu_f(s);
    }
    __syncthreads();
    float s = 0.f;                                 // lin2 [500]->[1]
    for (int k = t; k < LDIM; k += blockDim.x) s += y1[k] * lin2_w[k];
    red[t] = s; __syncthreads();
    for (int st = blockDim.x >> 1; st > 0; st >>= 1) {
        if (t < st) red[t] += red[t + st];
        __syncthreads();
    }
    if (t == 0) {
        float y = lrelu_f(red[0] + lin2_b[0]);
        out[0] = y;
        if (n_out > 1) out[1] = 0.0f;              // argmax of 1-logit softmax
    }
}

// ---------------------------------------------------------------------------
extern "C" void kernel_launch(void* const* d_in, const int* in_sizes, int n_in,
                              void* d_out, int out_size, void* d_ws, size_t ws_size,
                              hipStream_t stream)
{
    (void)n_in; (void)ws_size;
    const float* x        = (const float*)d_in[0];
    const float* conv1_w  = (const float*)d_in[1];
    const float* conv1_b  = (const float*)d_in[2];
    const float* conv2_w  = (const float*)d_in[3];
    const float* conv2_b  = (const float*)d_in[4];
    const float* fc_w     = (const float*)d_in[5];
    const float* fc_b     = (const float*)d_in[6];
    const float* se_wrel  = (const float*)d_in[7];
    const float* se_brel  = (const float*)d_in[8];
    const float* se_wroot = (const float*)d_in[9];
    const float* sp_wrel  = (const float*)d_in[10];
    const float* sp_brel  = (const float*)d_in[11];
    const float* sp_wroot = (const float*)d_in[12];
    const float* bn1_g    = (const float*)d_in[13];
    const float* bn1_b    = (const float*)d_in[14];
    const float* bn2_g    = (const float*)d_in[15];
    const float* bn2_b    = (const float*)d_in[16];
    const float* lin1_w   = (const float*)d_in[17];
    const float* lin1_b   = (const float*)d_in[18];
    const float* lin2_w   = (const float*)d_in[19];
    const float* lin2_b   = (const float*)d_in[20];
    float* out = (float*)d_out;

    const int n = in_sizes[0] / 784;               // 2048
    const int m = (int)((0.1 * n) + 0.999999);     // ceil(0.1*n) = 205

    // workspace layout (floats); conv1out region reused for Gram/Adjacency
    float* ws = (float*)d_ws;
    size_t o = 0;
    float* conv1out = ws + o; o += (size_t)n * 2880;   // >= n*n for n<=2880
    float* G        = conv1out;                        // reuse after conv2
    float* fcin     = ws + o; o += (size_t)n * 800;
    float* H        = ws + o; o += (size_t)n * LDIM;
    float* AH       = ws + o; o += (size_t)n * LDIM;   // becomes AGG in place
    float* Zpre     = ws + o; o += (size_t)n * LDIM;   // becomes Z in place
    float* sq       = ws + o; o += n;
    float* deg      = ws + o; o += n;
    unsigned int* hist = (unsigned int*)(ws + o); o += NBINS;
    float* thr      = ws + o; o += 4;
    float* meanZ    = ws + o; o += LDIM;
    float* varZ     = ws + o; o += LDIM;
    float* meanS    = ws + o; o += 2;
    float* varS     = ws + o; o += 2;
    float* Sb       = ws + o; o += (size_t)n * 2;
    float* U        = ws + o; o += (size_t)n * 2;
    float* Xp       = ws + o; o += 2 * LDIM;
    float* Ap       = ws + o; o += 4;

    const int nt16 = (LDIM + 15) / 16;                 // 32 N-tiles for 500
    dim3 gblk(128);

    // feature extractor
    conv1_pool<<<n, 256, 0, stream>>>(x, conv1_w, conv1_b, conv1out);
    conv2_pool<<<n, 256, 0, stream>>>(conv1out, conv2_w, conv2_b, fcin);
    // H = relu(fcin @ fc_w^T + fc_b)
    wmma_gemm_f32<1><<<dim3(nt16, n / 64), gblk, 0, stream>>>(
        fcin, fc_w, H, LDIM, 800, 800, 800, LDIM, fc_b, 0, 1);

    // adjacency: Gram, histogram-quantized chamfer threshold, A (in place)
    rowsq<<<(n + 255) / 256, 256, 0, stream>>>(H, sq, n);
    wmma_gemm_f32<1><<<dim3(n / 16, n / 64), gblk, 0, stream>>>(
        H, H, G, n, LDIM, LDIM, LDIM, n, nullptr, 0, 0);
    zero_u32<<<(NBINS + 255) / 256, 256, 0, stream>>>(hist, NBINS);
    hist_kernel<<<4096, 256, 0, stream>>>(G, sq, hist, n);
    thr_kernel<<<1, 64, 0, stream>>>(hist, thr, m);
    build_adj<<<n, 256, 0, stream>>>(G, sq, thr, deg, n);

    // AH = A @ H, then AGG = AH / deg (in place)
    wmma_gemm_f32<0><<<dim3(nt16, n / 64), gblk, 0, stream>>>(
        G, H, AH, LDIM, n, n, LDIM, LDIM, nullptr, 0, 0);
    {
        long total = (long)n * LDIM;
        scale_rows<<<(unsigned)((total + 255) / 256), 256, 0, stream>>>(AH, deg, total);
    }

    // Z = BN(lrelu(AGG @ Wrel^T + brel + H @ Wroot^T))
    wmma_gemm_f32<1><<<dim3(nt16, n / 64), gblk, 0, stream>>>(
        AH, se_wrel, Zpre, LDIM, LDIM, LDIM, LDIM, LDIM, se_brel, 0, 0);
    wmma_gemm_f32<1><<<dim3(nt16, n / 64), gblk, 0, stream>>>(
        H, se_wroot, Zpre, LDIM, LDIM, LDIM, LDIM, LDIM, nullptr, 1, 0);
    colstats<<<LDIM, 256, 0, stream>>>(Zpre, meanZ, varZ, n, LDIM);
    {
        long total = (long)n * LDIM;
        bn_apply<<<(unsigned)((total + 255) / 256), 256, 0, stream>>>(
            Zpre, meanZ, varZ, bn1_g, bn1_b, total, LDIM);
    }

    // S = softmax(softmax(BN(lrelu(AGG @ Wp^T + bp + H @ Wproot^T))))
    s_pre<<<(n + 255) / 256, 256, 0, stream>>>(AH, H, sp_wrel, sp_brel, sp_wroot, Sb, n);
    colstats<<<2, 256, 0, stream>>>(Sb, meanS, varS, n, 2);
    s_final<<<(n + 255) / 256, 256, 0, stream>>>(Sb, meanS, varS, bn2_g, bn2_b, n);

    // pooling: Xp = S^T Z, Ap = S^T A S
    a_times_s<<<n, 256, 0, stream>>>(G, Sb, U, n);
    pool_x<<<LDIM, 256, 0, stream>>>(Sb, Zpre, Xp, n);
    pool_a<<<1, 256, 0, stream>>>(Sb, U, Ap, n);

    // pooled SAGE + BN + lin1 + lin2
    head_kernel<<<1, 512, 0, stream>>>(Xp, Ap, se_wrel, se_brel, se_wroot,
                                       bn1_g, bn1_b, lin1_w, lin1_b, lin2_w, lin2_b,
                                       out, out_size);
}